// B4StemGCN_63385127355013
// MI455X (gfx1250) — compile-verified
//
#include <hip/hip_runtime.h>
#include <hip/hip_bf16.h>

typedef __attribute__((ext_vector_type(2))) float v2f;
typedef __attribute__((ext_vector_type(8))) float v8f;

namespace {
constexpr int kB = 64, kC = 64, kT = 300, kV = 25, kK = 3;
constexpr int kVP = 32;       // V padded to two 16-wide WMMA tiles
constexpr int kAStride = 33;  // LDS row stride for 32-wide tiles (bank-spread)
constexpr int kWStride = 65;  // LDS row stride for 64-wide W rows (conflict-free)
constexpr int kWaves = 8;
constexpr int kWGs = 300;
constexpr int kBT = kB * kT;                          // 19200 (b,t) tiles
constexpr int kItersPerWave = kBT / (kWGs * kWaves);  // 8
constexpr float kEps = 1e-5f;
constexpr float kInvK = 1.0f / 3.0f;
}  // namespace

__global__ void k_zero_stats(float* stats) { stats[threadIdx.x] = 0.0f; }

// Phase 1: fused  Y_bt = sum_k (W_k/K) @ (X_bt @ A_eff_k)  via f32 WMMA.
// Writes pre-BN y to workspace; accumulates per-channel sum / sumsq.
// Bias dropped: a per-channel constant shift is exactly cancelled by BN.
__global__ __launch_bounds__(kWaves * 32, 1) void k_gcn_gemm(
    const float* __restrict__ x, const float* __restrict__ A_fixed,
    const float* __restrict__ A_edge, const float* __restrict__ W,
    float* __restrict__ y_pre, float* __restrict__ stats) {
  extern __shared__ float smem[];
  float* A_s = smem;                       // [3][32][33]  padded A_fixed*A_edge
  float* W_s = A_s + kK * kVP * kAStride;  // [3][64][65]  W / K
  float* st_s = W_s + kK * kC * kWStride;  // [128]        per-WG sum/sumsq
  float* XA_s = st_s + 128;                // [8 waves][64][33] stage-A spill

  const int tid = threadIdx.x;
  const int lane = tid & 31;
  const int wave = tid >> 5;
  const int llo = lane & 15;    // column-within-tile lane id
  const int lhi = lane >> 4;    // lane half (WMMA fragment layout)
  float* xa = XA_s + wave * (kC * kAStride);

  // ---- one-time LDS staging (shared by all 8 waves, all 8 tiles) ----
  for (int i = tid; i < kK * kVP * kVP; i += blockDim.x) {
    const int k = i >> 10, r = (i >> 5) & 31, c = i & 31;
    float v = 0.0f;
    if (r < kV && c < kV) {
      const int g = (k * kV + r) * kV + c;
      v = A_fixed[g] * A_edge[g];
    }
    A_s[(k * kVP + r) * kAStride + c] = v;
  }
  for (int i = tid; i < kK * kC * kC; i += blockDim.x) {
    const int k = i / (kC * kC), rem = i % (kC * kC);
    W_s[(k * kC + rem / kC) * kWStride + (rem % kC)] = W[i] * kInvK;
  }
  for (int i = tid; i < 128; i += blockDim.x) st_s[i] = 0.0f;
  __syncthreads();

  const v8f vzero = {};
  const int slot = blockIdx.x * kWaves + wave;

  for (int it = 0; it < kItersPerWave; ++it) {
    const int bt = slot * kItersPerWave + it;
    const int b = bt / kT, t = bt % kT;
    const float* xb = x + ((size_t)b * kC * kT + t) * (size_t)kV;

    // Prefetch next tile's rows (2 rows per lane -> 64 rows).
    if (it + 1 < kItersPerWave) {
      const int bt2 = bt + 1;
      const float* nb =
          x + ((size_t)(bt2 / kT) * kC * kT + (bt2 % kT)) * (size_t)kV;
      __builtin_prefetch(nb + (size_t)(2 * lane) * (kT * kV), 0, 0);
      __builtin_prefetch(nb + (size_t)(2 * lane + 1) * (kT * kV), 0, 0);
    }

    // ---- X as WMMA A-fragments, loaded once, reused for all 3 branches ----
    // A 16x4 f32 frag: v0 = X[M=llo][K=2*lhi], v1 = X[M=llo][K=2*lhi+1]
    v2f xf[4][8];
#pragma unroll
    for (int mt = 0; mt < 4; ++mt) {
      const float* xr = xb + (size_t)(mt * 16 + llo) * (kT * kV);
#pragma unroll
      for (int kg = 0; kg < 8; ++kg) {
        const int v0 = kg * 4 + 2 * lhi;
        xf[mt][kg].x = (v0 < kV) ? xr[v0] : 0.0f;
        xf[mt][kg].y = (v0 + 1 < kV) ? xr[v0 + 1] : 0.0f;
      }
    }

    v8f yac[4][2];
#pragma unroll
    for (int mt = 0; mt < 4; ++mt)
#pragma unroll
      for (int nt = 0; nt < 2; ++nt) yac[mt][nt] = vzero;

    for (int k = 0; k < kK; ++k) {
      // ---- stage A: XA = X(64x32) @ A_k(32x32) ----
      v8f xaacc[4][2];
#pragma unroll
      for (int mt = 0; mt < 4; ++mt)
#pragma unroll
        for (int nt = 0; nt < 2; ++nt) xaacc[mt][nt] = vzero;

      const float* Ak = A_s + k * kVP * kAStride;
#pragma unroll
      for (int kg = 0; kg < 8; ++kg) {
        const int brow = kg * 4 + 2 * lhi;
        v2f bf[2];
#pragma unroll
        for (int nt = 0; nt < 2; ++nt) {
          const float* bp = Ak + brow * kAStride + nt * 16 + llo;
          bf[nt].x = bp[0];
          bf[nt].y = bp[kAStride];
        }
#pragma unroll
        for (int mt = 0; mt < 4; ++mt)
#pragma unroll
          for (int nt = 0; nt < 2; ++nt)
            xaacc[mt][nt] = __builtin_amdgcn_wmma_f32_16x16x4_f32(
                false, xf[mt][kg], false, bf[nt], (short)0, xaacc[mt][nt],
                false, false);
      }

      // ---- C/D-layout -> row-major spill (wave-private, DS in-order) ----
      __builtin_amdgcn_wave_barrier();
#pragma unroll
      for (int mt = 0; mt < 4; ++mt)
#pragma unroll
        for (int nt = 0; nt < 2; ++nt)
#pragma unroll
          for (int g = 0; g < 8; ++g)
            xa[(mt * 16 + g + 8 * lhi) * kAStride + nt * 16 + llo] =
                xaacc[mt][nt][g];
      __builtin_amdgcn_wave_barrier();

      // ---- stage B: yac += (W_k/K)(64x64) @ XA(64x32) ----
      const float* Wk = W_s + k * kC * kWStride;
      for (int kk = 0; kk < kC; kk += 4) {
        const int cb = kk + 2 * lhi;
        v2f bf[2];
#pragma unroll
        for (int nt = 0; nt < 2; ++nt) {
          const float* bp = xa + cb * kAStride + nt * 16 + llo;
          bf[nt].x = bp[0];
          bf[nt].y = bp[kAStride];
        }
#pragma unroll
        for (int mt = 0; mt < 4; ++mt) {
          const float* ap = Wk + (mt * 16 + llo) * kWStride + cb;
          v2f af;
          af.x = ap[0];
          af.y = ap[1];
#pragma unroll
          for (int nt = 0; nt < 2; ++nt)
            yac[mt][nt] = __builtin_amdgcn_wmma_f32_16x16x4_f32(
                false, af, false, bf[nt], (short)0, yac[mt][nt], false, false);
        }
      }
    }

    // ---- store pre-BN y, accumulate per-channel stats ----
#pragma unroll
    for (int mt = 0; mt < 4; ++mt)
#pragma unroll
      for (int g = 0; g < 8; ++g) {
        const int o = mt * 16 + g + 8 * lhi;
        float* yo = y_pre + (((size_t)b * kC + o) * kT + t) * (size_t)kV;
        float s = 0.0f, sq = 0.0f;
#pragma unroll
        for (int nt = 0; nt < 2; ++nt) {
          const int w = nt * 16 + llo;
          const float v = yac[mt][nt][g];
          if (w < kV) {
            yo[w] = v;
            s += v;
            sq += v * v;
          }
        }
        // Reduce across the 16 lanes of this half (all share channel o).
#pragma unroll
        for (int off = 1; off < 16; off <<= 1) {
          s += __shfl_xor(s, off, 32);
          sq += __shfl_xor(sq, off, 32);
        }
        if (llo == 0) {
          atomicAdd(st_s + o, s);
          atomicAdd(st_s + 64 + o, sq);
        }
      }
  }

  __syncthreads();
  for (int i = tid; i < 128; i += blockDim.x) atomicAdd(stats + i, st_s[i]);
}

// Phase 2: BatchNorm (training stats from phase 1) + residual + SiLU.
__global__ __launch_bounds__(256) void k_bn_silu(
    const float* __restrict__ y_pre, const float* __restrict__ x,
    const float* __restrict__ stats, const float* __restrict__ gamma,
    const float* __restrict__ beta, float* __restrict__ out) {
  constexpr size_t N = (size_t)kB * kC * kT * kV;  // 30,720,000 (div by 4)
  constexpr float invN = 1.0f / (float)(kB * kT * kV);
  const size_t stride = (size_t)gridDim.x * blockDim.x * 4;
  for (size_t i = ((size_t)blockIdx.x * blockDim.x + threadIdx.x) * 4; i < N;
       i += stride) {
    const float4 yp = *(const float4*)(y_pre + i);
    const float4 xv = *(const float4*)(x + i);
    const float r[4] = {yp.x, yp.y, yp.z, yp.w};
    const float xr[4] = {xv.x, xv.y, xv.z, xv.w};
    float o4[4];
#pragma unroll
    for (int j = 0; j < 4; ++j) {
      const int o = (int)((i + j) / (kT * kV)) % kC;
      const float mean = stats[o] * invN;
      const float var = fmaxf(stats[64 + o] * invN - mean * mean, 0.0f);
      const float yn = (r[j] - mean) * rsqrtf(var + kEps);
      const float z = yn * gamma[o] + beta[o] + xr[j];
      o4[j] = z / (1.0f + __expf(-z));
    }
    *(float4*)(out + i) = make_float4(o4[0], o4[1], o4[2], o4[3]);
  }
}

extern "C" void kernel_launch(void* const* d_in, const int* in_sizes, int n_in,
                              void* d_out, int out_size, void* d_ws,
                              size_t ws_size, hipStream_t stream) {
  const float* x = (const float*)d_in[0];
  const float* A_fixed = (const float*)d_in[1];
  const float* A_edge = (const float*)d_in[2];
  const float* W = (const float*)d_in[3];
  // d_in[4] = per-branch bias: per-channel constant, exactly cancelled by the
  // BatchNorm mean subtraction -> intentionally unused.
  const float* gamma = (const float*)d_in[5];
  const float* beta = (const float*)d_in[6];
  float* out = (float*)d_out;

  float* stats = (float*)d_ws;   // [64] sum, [64] sumsq
  float* y_pre = stats + 128;    // [B,C,T,V] pre-BN activations

  k_zero_stats<<<1, 128, 0, stream>>>(stats);

  const size_t smem_bytes =
      (size_t)(kK * kVP * kAStride + kK * kC * kWStride + 128 +
               kWaves * kC * kAStride) *
      sizeof(float);  // ~128 KB / WG (dynamic LDS, within 320 KB WGP budget)
  k_gcn_gemm<<<kWGs, kWaves * 32, smem_bytes, stream>>>(x, A_fixed, A_edge, W,
                                                        y_pre, stats);

  k_bn_silu<<<7500, 256, 0, stream>>>(y_pre, x, stats, gamma, beta, out);
}